// GraphSage_50311246905373
// MI455X (gfx1250) — compile-verified
//
#include <hip/hip_runtime.h>

#define N_NODES 50000
#define N_EDGES 400000
#define F_IN    256
#define HID     1024
#define F_MID   512
#define OUT_DIM 10
#define EPS_BN  1e-5f

typedef __attribute__((ext_vector_type(2))) float v2f;
typedef __attribute__((ext_vector_type(8))) float v8f;

// ---------------------------------------------------------------------------
// Degree count: deg[dst] += 1 per edge
// ---------------------------------------------------------------------------
__global__ void count_deg(const int* __restrict__ ei, float* __restrict__ deg, int E) {
    int e = blockIdx.x * blockDim.x + threadIdx.x;
    if (e < E) atomicAdd(&deg[ei[E + e]], 1.0f);
}

// ---------------------------------------------------------------------------
// Scatter-add aggregation: one wave per edge, lanes stripe the feature dim.
// agg[dst, :] += X[src, :]
// ---------------------------------------------------------------------------
__global__ void scatter_accum(const float* __restrict__ X, const int* __restrict__ ei,
                              float* __restrict__ agg, int E, int F) {
    int wv = threadIdx.x >> 5, lane = threadIdx.x & 31;
    int e = blockIdx.x * (blockDim.x >> 5) + wv;
    if (e >= E) return;
    int s = ei[e], d = ei[E + e];
    const float* xp = X + (size_t)s * F;
    float* ap = agg + (size_t)d * F;
    for (int f = lane; f < F; f += 32) atomicAdd(ap + f, xp[f]);
}

// agg[r, :] /= max(deg[r], 1)
__global__ void mean_div(float* __restrict__ agg, const float* __restrict__ deg,
                         long long total, int F) {
    long long idx = (long long)blockIdx.x * blockDim.x + threadIdx.x;
    if (idx >= total) return;
    int r = (int)(idx / F);
    agg[idx] *= (1.0f / fmaxf(deg[r], 1.0f));
}

// ---------------------------------------------------------------------------
// Fused WMMA fp32 GEMM: C = A@W [+ A2@W2] + bias [+ReLU]
//   A:[M,K] row-major, W:[K,Fout] row-major, C:[M,Fout]
// Block = 128 threads = 4 waves; each wave computes a 16x64 tile of C using
// V_WMMA_F32_16X16X4_F32, 4 accumulators of 16x16, K-step 4.
// A fragment (16x4, 2 VGPRs):  lane<16 -> row=lane, (K0,K1); lane>=16 -> (K2,K3)
// B fragment (4x16, 2 VGPRs):  lane<16 -> col=lane, rows(K0,K1); lane>=16 -> rows(K2,K3)
// C/D: VGPR v -> M = v + 8*(lane>=16), N = lane&15
// ---------------------------------------------------------------------------
__global__ void gemm_wmma_f32(const float* __restrict__ A,  const float* __restrict__ W,
                              const float* __restrict__ A2, const float* __restrict__ W2,
                              const float* __restrict__ bias, float* __restrict__ C,
                              int M, int K, int Fout, int relu) {
    const int lane  = threadIdx.x & 31;
    const int wv    = threadIdx.x >> 5;
    const int mtile = blockIdx.x * 4 + wv;
    if (mtile * 16 >= M) return;               // wave-uniform guard (EXEC stays full)

    const int hi    = lane >> 4;               // 0 or 1 (half-wave)
    const int lo    = lane & 15;
    const int row   = mtile * 16 + lo;         // A row this lane feeds
    const int nbase = blockIdx.y * 64;

    v8f acc0 = {}, acc1 = {}, acc2 = {}, acc3 = {};

    const float* Aptr  = A  + (size_t)row * K;
    const float* A2ptr = A2 ? A2 + (size_t)row * K : nullptr;

    for (int k = 0; k < K; k += 4) {
        v2f a;
        a.x = Aptr[k + 2 * hi];
        a.y = Aptr[k + 2 * hi + 1];
        const float* w0 = W + (size_t)(k + 2 * hi) * Fout + nbase + lo;
        const float* w1 = w0 + Fout;
        v2f b0; b0.x = w0[0];  b0.y = w1[0];
        v2f b1; b1.x = w0[16]; b1.y = w1[16];
        v2f b2; b2.x = w0[32]; b2.y = w1[32];
        v2f b3; b3.x = w0[48]; b3.y = w1[48];
        acc0 = __builtin_amdgcn_wmma_f32_16x16x4_f32(false, a, false, b0, (short)0, acc0, false, false);
        acc1 = __builtin_amdgcn_wmma_f32_16x16x4_f32(false, a, false, b1, (short)0, acc1, false, false);
        acc2 = __builtin_amdgcn_wmma_f32_16x16x4_f32(false, a, false, b2, (short)0, acc2, false, false);
        acc3 = __builtin_amdgcn_wmma_f32_16x16x4_f32(false, a, false, b3, (short)0, acc3, false, false);
    }

    if (A2ptr) {
        for (int k = 0; k < K; k += 4) {
            v2f a;
            a.x = A2ptr[k + 2 * hi];
            a.y = A2ptr[k + 2 * hi + 1];
            const float* w0 = W2 + (size_t)(k + 2 * hi) * Fout + nbase + lo;
            const float* w1 = w0 + Fout;
            v2f b0; b0.x = w0[0];  b0.y = w1[0];
            v2f b1; b1.x = w0[16]; b1.y = w1[16];
            v2f b2; b2.x = w0[32]; b2.y = w1[32];
            v2f b3; b3.x = w0[48]; b3.y = w1[48];
            acc0 = __builtin_amdgcn_wmma_f32_16x16x4_f32(false, a, false, b0, (short)0, acc0, false, false);
            acc1 = __builtin_amdgcn_wmma_f32_16x16x4_f32(false, a, false, b1, (short)0, acc1, false, false);
            acc2 = __builtin_amdgcn_wmma_f32_16x16x4_f32(false, a, false, b2, (short)0, acc2, false, false);
            acc3 = __builtin_amdgcn_wmma_f32_16x16x4_f32(false, a, false, b3, (short)0, acc3, false, false);
        }
    }

    // Epilogue: VGPR v holds (M = mtile*16 + v + 8*hi, N = nbase + 16*t + lo)
    v8f accs[4] = {acc0, acc1, acc2, acc3};
#pragma unroll
    for (int t = 0; t < 4; ++t) {
        int c = nbase + 16 * t + lo;
        float bv = bias[c];
#pragma unroll
        for (int v = 0; v < 8; ++v) {
            int r = mtile * 16 + v + 8 * hi;
            float val = accs[t][v] + bv;
            if (relu) val = fmaxf(val, 0.0f);
            C[(size_t)r * Fout + c] = val;
        }
    }
}

// ---------------------------------------------------------------------------
// BatchNorm (training mode, over node dim)
// ---------------------------------------------------------------------------
__global__ void colstat(const float* __restrict__ Z, float* __restrict__ sum,
                        float* __restrict__ sumsq, int Nrows, int F, int rowsPerBlock) {
    int c  = blockIdx.x * blockDim.x + threadIdx.x;   // F is a multiple of blockDim
    int r0 = blockIdx.y * rowsPerBlock;
    int r1 = min(r0 + rowsPerBlock, Nrows);
    float s = 0.0f, q = 0.0f;
    for (int r = r0; r < r1; ++r) {
        float v = Z[(size_t)r * F + c];
        s += v; q += v * v;
    }
    atomicAdd(&sum[c], s);
    atomicAdd(&sumsq[c], q);
}

// Turn (sum, sumsq) into (scale, shift) in place
__global__ void bn_finalize(float* __restrict__ sum, float* __restrict__ sumsq,
                            const float* __restrict__ g, const float* __restrict__ be,
                            int Nrows, int F) {
    int c = blockIdx.x * blockDim.x + threadIdx.x;
    if (c >= F) return;
    float inv = 1.0f / (float)Nrows;
    float m   = sum[c] * inv;
    float var = sumsq[c] * inv - m * m;
    float scale = g[c] * rsqrtf(var + EPS_BN);
    sum[c]   = scale;
    sumsq[c] = be[c] - m * scale;
}

// Z = relu(Z*scale + shift), in place
__global__ void bn_apply_relu(float* __restrict__ Z, const float* __restrict__ scale,
                              const float* __restrict__ shift, long long total, int F) {
    long long idx = (long long)blockIdx.x * blockDim.x + threadIdx.x;
    if (idx >= total) return;
    int c = (int)(idx % F);
    float v = Z[idx] * scale[c] + shift[c];
    Z[idx] = fmaxf(v, 0.0f);
}

// ---------------------------------------------------------------------------
// Skinny head GEMM: out[M,10] = H[M,512] @ Wo + bo  (too narrow for WMMA tiles)
// ---------------------------------------------------------------------------
__global__ void head_gemm(const float* __restrict__ H, const float* __restrict__ Wo,
                          const float* __restrict__ bo, float* __restrict__ out,
                          int M, int K, int Nout) {
    int idx = blockIdx.x * blockDim.x + threadIdx.x;
    if (idx >= M * Nout) return;
    int r = idx / Nout, c = idx % Nout;
    const float* hp = H + (size_t)r * K;
    float s = bo[c];
    for (int k = 0; k < K; ++k) s += hp[k] * Wo[(size_t)k * Nout + c];
    out[idx] = s;
}

// ---------------------------------------------------------------------------
extern "C" void kernel_launch(void* const* d_in, const int* in_sizes, int n_in,
                              void* d_out, int out_size, void* d_ws, size_t ws_size,
                              hipStream_t stream) {
    (void)in_sizes; (void)n_in; (void)out_size; (void)ws_size;

    const float* x   = (const float*)d_in[0];
    const int*   ei  = (const int*)  d_in[1];
    const float* W1l = (const float*)d_in[2];
    const float* b1  = (const float*)d_in[3];
    const float* W1r = (const float*)d_in[4];
    const float* g1  = (const float*)d_in[5];
    const float* be1 = (const float*)d_in[6];
    const float* W2l = (const float*)d_in[7];
    const float* b2  = (const float*)d_in[8];
    const float* W2r = (const float*)d_in[9];
    const float* g2  = (const float*)d_in[10];
    const float* be2 = (const float*)d_in[11];
    const float* Wf  = (const float*)d_in[12];
    const float* bf  = (const float*)d_in[13];
    const float* Wo  = (const float*)d_in[14];
    const float* bo  = (const float*)d_in[15];
    float* out = (float*)d_out;

    // ---- workspace carve-up (256B aligned) ----
    char* ws = (char*)d_ws;
    size_t off = 0;
    auto carve = [&](size_t bytes) -> void* {
        void* p = ws + off;
        off = (off + bytes + 255) & ~(size_t)255;
        return p;
    };
    float* deg   = (float*)carve((size_t)N_NODES * sizeof(float));
    float* stat0 = (float*)carve((size_t)HID * sizeof(float));      // sum  -> scale
    float* stat1 = (float*)carve((size_t)HID * sizeof(float));      // sumsq-> shift
    float* agg   = (float*)carve((size_t)N_NODES * HID * sizeof(float));  // reused both layers
    float* h1    = (float*)carve((size_t)N_NODES * HID * sizeof(float));  // z1/h1, later h3
    float* z2    = (float*)carve((size_t)N_NODES * HID * sizeof(float));  // z2/h2
    float* h3    = h1;  // h1 dead after layer-2 BN; reuse for [N, F_MID]

    const int THR = 256;

    // ---- degree counts ----
    hipMemsetAsync(deg, 0, (size_t)N_NODES * sizeof(float), stream);
    count_deg<<<(N_EDGES + THR - 1) / THR, THR, 0, stream>>>(ei, deg, N_EDGES);

    // ---- layer 1: mean aggregation of x (F_IN feats) ----
    hipMemsetAsync(agg, 0, (size_t)N_NODES * F_IN * sizeof(float), stream);
    scatter_accum<<<(N_EDGES * 32 + THR - 1) / THR, THR, 0, stream>>>(x, ei, agg, N_EDGES, F_IN);
    {
        long long total = (long long)N_NODES * F_IN;
        mean_div<<<(int)((total + THR - 1) / THR), THR, 0, stream>>>(agg, deg, total, F_IN);
    }

    // ---- layer 1 fused GEMM: z1 = agg@W1l + x@W1r + b1 ----
    {
        dim3 grid((N_NODES / 16 + 3) / 4, HID / 64);
        gemm_wmma_f32<<<grid, 128, 0, stream>>>(agg, W1l, x, W1r, b1, h1,
                                                N_NODES, F_IN, HID, /*relu=*/0);
    }

    // ---- BN1 + ReLU (in place on h1) ----
    hipMemsetAsync(stat0, 0, HID * sizeof(float), stream);
    hipMemsetAsync(stat1, 0, HID * sizeof(float), stream);
    {
        dim3 grid(HID / THR, (N_NODES + 255) / 256);
        colstat<<<grid, THR, 0, stream>>>(h1, stat0, stat1, N_NODES, HID, 256);
        bn_finalize<<<(HID + THR - 1) / THR, THR, 0, stream>>>(stat0, stat1, g1, be1, N_NODES, HID);
        long long total = (long long)N_NODES * HID;
        bn_apply_relu<<<(int)((total + THR - 1) / THR), THR, 0, stream>>>(h1, stat0, stat1, total, HID);
    }

    // ---- layer 2: mean aggregation of h1 (HID feats) ----
    hipMemsetAsync(agg, 0, (size_t)N_NODES * HID * sizeof(float), stream);
    scatter_accum<<<(N_EDGES * 32 + THR - 1) / THR, THR, 0, stream>>>(h1, ei, agg, N_EDGES, HID);
    {
        long long total = (long long)N_NODES * HID;
        mean_div<<<(int)((total + THR - 1) / THR), THR, 0, stream>>>(agg, deg, total, HID);
    }

    // ---- layer 2 fused GEMM: z2 = agg@W2l + h1@W2r + b2 ----
    {
        dim3 grid((N_NODES / 16 + 3) / 4, HID / 64);
        gemm_wmma_f32<<<grid, 128, 0, stream>>>(agg, W2l, h1, W2r, b2, z2,
                                                N_NODES, HID, HID, /*relu=*/0);
    }

    // ---- BN2 + ReLU (in place on z2) ----
    hipMemsetAsync(stat0, 0, HID * sizeof(float), stream);
    hipMemsetAsync(stat1, 0, HID * sizeof(float), stream);
    {
        dim3 grid(HID / THR, (N_NODES + 255) / 256);
        colstat<<<grid, THR, 0, stream>>>(z2, stat0, stat1, N_NODES, HID, 256);
        bn_finalize<<<(HID + THR - 1) / THR, THR, 0, stream>>>(stat0, stat1, g2, be2, N_NODES, HID);
        long long total = (long long)N_NODES * HID;
        bn_apply_relu<<<(int)((total + THR - 1) / THR), THR, 0, stream>>>(z2, stat0, stat1, total, HID);
    }

    // ---- FC + ReLU: h3 = relu(h2 @ Wf + bf) ----
    {
        dim3 grid((N_NODES / 16 + 3) / 4, F_MID / 64);
        gemm_wmma_f32<<<grid, 128, 0, stream>>>(z2, Wf, nullptr, nullptr, bf, h3,
                                                N_NODES, HID, F_MID, /*relu=*/1);
    }

    // ---- head: out = h3 @ Wo + bo ----
    {
        int total = N_NODES * OUT_DIM;
        head_gemm<<<(total + THR - 1) / THR, THR, 0, stream>>>(h3, Wo, bo, out,
                                                               N_NODES, F_MID, OUT_DIM);
    }
}